// ResultsGenerationModule_26079041421793
// MI455X (gfx1250) — compile-verified
//
#include <hip/hip_runtime.h>
#include <math.h>

// Problem constants (from reference)
#define NB       2
#define CIN      256
#define HH       160
#define WW       160
#define HWPLANE  (HH * WW)          // 25600
#define NPTS     9
#define BINS     8
#define KTOP     4
#define CLS      80
#define OREG     144                // NP*2*BINS
#define OTOT     224                // CLS + OREG
#define CONF_CH  64
#define GROUPS   8
#define CH_PER_G (CONF_CH / GROUPS) // 8
#define GN_EPS   1e-5f

#define TILE_M   32                 // pixels per block
#define NTILES   14                 // 5 cls tiles + 9 reg tiles
#define NTASKS   (2 * NTILES)       // 2 M-subtiles x 14 N-tiles = 28
#define A_STRIDE 40                 // 32 data dwords + 8 pad dwords per K row

typedef float v2f __attribute__((ext_vector_type(2)));
typedef float v8f __attribute__((ext_vector_type(8)));
typedef unsigned int v4u __attribute__((ext_vector_type(4)));
typedef int v8i __attribute__((ext_vector_type(8)));
typedef int v4i __attribute__((ext_vector_type(4)));

__device__ __forceinline__ float sigmoidf(float x) {
    return 1.0f / (1.0f + __expf(-x));
}

#if __has_builtin(__builtin_amdgcn_tensor_load_to_lds)
// TDM DMA: 2D tile [CIN rows x TILE_M floats], row stride = HWPLANE elems,
// into LDS with 8 pad dwords inserted every 32 dwords (A_STRIDE=40 layout).
// D# per CDNA5 ISA ch.8 (group0: count/lds_addr/global_addr/type=2;
// group1: data_size=4B, pad_enable, pad_interval=32dw, pad_amount=8dw,
// tensor_dim0=HW, tensor_dim1=CIN, tile=32x256, dim0_stride=HW).
// This toolchain exposes the 6-arg builtin:
//   (uint32x4 g0, int32x8 g1, int32x4 g2, int32x4 g3, int32x8 g4, i32 cpol)
__device__ __forceinline__ void tdm_stage_tile(unsigned lds_addr,
                                               const float* gsrc)
{
    const unsigned long long ga = (unsigned long long)(size_t)gsrc;
    v4u g0;
    g0[0] = 1u;                                   // count=1, user mode
    g0[1] = lds_addr;                             // lds_addr [63:32]
    g0[2] = (unsigned)(ga & 0xffffffffull);       // global_addr lo
    g0[3] = (unsigned)((ga >> 32) & 0x1ffffffull) // global_addr hi (57-bit)
          | (2u << 30);                           // type=2 ("image")
    v8i g1;
    g1[0] = (int)((2u << 16)                      // data_size = 4 bytes
          | (1u << 20)                            // pad_enable
          | (4u << 22)                            // pad_interval: 32 dwords
          | (7u << 25));                          // pad_amount : 8 dwords
    g1[1] = (int)((HWPLANE & 0xffff) << 16);      // tensor_dim0 lo16
    g1[2] = (int)(((HWPLANE >> 16) & 0xffff)      // tensor_dim0 hi16
          | ((CIN & 0xffff) << 16));              // tensor_dim1 lo16
    g1[3] = (int)((TILE_M & 0xffff) << 16);       // tile_dim0 = 32
    g1[4] = (int)(CIN & 0xffff);                  // tile_dim1 = 256
    g1[5] = (int)HWPLANE;                         // tensor_dim0_stride lo32
    g1[6] = 0;                                    // stride0 hi16 | stride1 lo16
    g1[7] = 0;                                    // stride1 hi32
    v4i gz4 = {0, 0, 0, 0};                       // 2D tensor: groups 2/3 unused
    v8i gz8 = {0, 0, 0, 0, 0, 0, 0, 0};
    __builtin_amdgcn_tensor_load_to_lds(g0, g1, gz4, gz4, gz8, 0);
}
#endif

// ---------------------------------------------------------------------------
// Kernel A: fused dual 1x1-conv GEMM (WMMA f32 16x16x4) + parallel head
// grid = (HWPLANE/32, NB), block = 256 (8 waves of 32)
// ---------------------------------------------------------------------------
__global__ __launch_bounds__(256) void head_gemm_kernel(
    const float* __restrict__ cls_feat,   // [NB][CIN][HW]
    const float* __restrict__ reg_feat,   // [NB][CIN][HW]
    const float* __restrict__ pts_init,   // [NB][2*NPTS][HW]
    const float* __restrict__ scale_p,    // [1]
    const float* __restrict__ cls_w,      // [CLS][CIN]
    const float* __restrict__ cls_b,      // [CLS]
    const float* __restrict__ reg_w,      // [OREG][CIN]
    const float* __restrict__ reg_b,      // [OREG]
    const float* __restrict__ conf1_w,    // [CONF_CH][16]
    const float* __restrict__ conf1_b,    // [CONF_CH]
    float* __restrict__ out,              // cls-sigmoid region + bbox region
    float* __restrict__ hid_ws,           // [NB][CONF_CH][HW]
    float* __restrict__ part_ws)          // [NB][GROUPS][2]  (sum, sumsq)
{
    __shared__ __align__(16) float sA[2 * CIN * A_STRIDE]; // 80 KB padded A
    __shared__ float sL[TILE_M][OTOT + 9];                 // 233-word rows
    __shared__ float sPts[2][NPTS][TILE_M + 4];            // pts_pred [coord][ip][m]
    __shared__ float sT16[TILE_M][17];                     // top-k vector per pixel
    __shared__ float sRed[TILE_M][GROUPS][2];              // GN partials

    const int n     = blockIdx.y;
    const int pbase = blockIdx.x * TILE_M;
    const int tid   = threadIdx.x;
    const int lane  = tid & 31;
    const int wave  = tid >> 5;

    const float* cbase = cls_feat + (size_t)n * CIN * HWPLANE + pbase;
    const float* rbase = reg_feat + (size_t)n * CIN * HWPLANE + pbase;

#if __has_builtin(__builtin_amdgcn_tensor_load_to_lds)
    // ---- TDM: one DMA per feature map stages the whole padded A slab ----
    if (wave == 0) {
        const unsigned base = (unsigned)(size_t)&sA[0];
        tdm_stage_tile(base, cbase);
        tdm_stage_tile(base + CIN * A_STRIDE * 4u, rbase);
#if __has_builtin(__builtin_amdgcn_s_wait_tensorcnt)
        __builtin_amdgcn_s_wait_tensorcnt(0);
#else
        asm volatile("s_wait_tensorcnt 0x0" ::: "memory");
#endif
    }
#else
    // ---- fallback: cooperative staging into the same padded layout ----
    for (int e = tid; e < 2 * CIN * TILE_M; e += 256) {
        const int f = e / (CIN * TILE_M);
        const int r = e - f * (CIN * TILE_M);
        const int k = r / TILE_M;
        const int m = r - k * TILE_M;
        const float* src = f ? rbase : cbase;
        sA[(f * CIN + k) * A_STRIDE + m] = src[(size_t)k * HWPLANE + m];
    }
#endif
    __syncthreads();

    // ---- WMMA GEMM: D[16 pix][16 ch] += A[16x4] * B[4x16] over K=256 ----
    // A lane layout (ISA): m = lane&15, VGPR0:k=2*(lane>>4), VGPR1:k+1
    // B lane layout      : n = lane&15, VGPR0:k=2*(lane>>4), VGPR1:k+1
    const int mfrag = lane & 15;
    const int koff  = (lane >> 4) * 2;

    for (int task = wave; task < NTASKS; task += 8) {
        const int   sm    = task & 1;         // M subtile (16 pixels)
        const int   t     = task >> 1;        // N tile
        const int   isReg = (t >= 5);
        const float* wmat = isReg ? reg_w : cls_w;
        const int   obase = isReg ? (t - 5) * 16 : t * 16;
        const int   ocol  = obase + (lane & 15);
        const float* sa   = &sA[isReg * CIN * A_STRIDE + sm * 16];

        v8f acc = {};
        for (int kk = 0; kk < CIN; kk += 4) {
            const int k = kk + koff;
            v2f a, b;
            a[0] = sa[k * A_STRIDE + mfrag];
            a[1] = sa[(k + 1) * A_STRIDE + mfrag];
            const float2 bw = *(const float2*)(wmat + (size_t)ocol * CIN + k);
            b[0] = bw.x;
            b[1] = bw.y;
            acc = __builtin_amdgcn_wmma_f32_16x16x4_f32(
                false, a, false, b, (short)0, acc, false, false);
        }

        // D layout (ISA): row m = r + 8*(lane>>4), col = lane&15
        const int mbase   = sm * 16 + (lane >> 4) * 8;
        const int colBase = (isReg ? CLS + obase : obase) + (lane & 15);
        #pragma unroll
        for (int r = 0; r < 8; ++r)
            sL[mbase + r][colBase] = acc[r];
    }
    __syncthreads();

    // ================= Phase A: all 256 threads =================
    // A1: per-(pixel, coord) softmax over 8 bins; probs back into sL;
    //     pts_pred into sPts.  c = task>>5 is wave-uniform.
    {
        const float scl = scale_p[0];
        for (int task = tid; task < 2 * NPTS * TILE_M; task += 256) {
            const int m = task & 31;
            const int c = task >> 5;
            float l[BINS];
            float mx = -1e30f;
            #pragma unroll
            for (int b = 0; b < BINS; ++b) {
                l[b] = (sL[m][CLS + c * BINS + b] + reg_b[c * BINS + b]) * scl;
                mx = fmaxf(mx, l[b]);
            }
            float s = 0.f;
            #pragma unroll
            for (int b = 0; b < BINS; ++b) { l[b] = __expf(l[b] - mx); s += l[b]; }
            const float inv = 1.f / s;
            float refine = 0.f;
            #pragma unroll
            for (int b = 0; b < BINS; ++b) {
                const float pr = l[b] * inv;
                sL[m][CLS + c * BINS + b] = pr;    // keep probs for phase B
                refine += pr * (float)b;
            }
            sPts[c & 1][c >> 1][m] =
                pts_init[((size_t)n * 2 * NPTS + c) * HWPLANE + pbase + m] + refine;
        }
        // A2: cls sigmoid, (pixel, channel) over all threads; o is
        //     wave-uniform, stores coalesced over m.
        for (int idx = tid; idx < CLS * TILE_M; idx += 256) {
            const int m = idx & 31;
            const int o = idx >> 5;
            out[((size_t)n * CLS + o) * HWPLANE + pbase + m] =
                sigmoidf(sL[m][o] + cls_b[o]);
        }
    }
    __syncthreads();

    // ================= Phase B: 128 threads, one (pixel, side) each ========
    // side g: 0=left(min x), 1=up(min y), 2=right(max x), 3=bottom(max y).
    // First-occurrence argmin/argmax (strict compare, ascending scan).
    if (tid < 4 * TILE_M) {
        const int m  = tid & 31;
        const int g  = tid >> 5;              // wave-uniform
        const int cb = (g & 1) ? 0 : 1;       // x for left/right, y for up/bottom
        float best = sPts[cb][0][m];
        int   bi   = 0;
        #pragma unroll
        for (int i = 1; i < NPTS; ++i) {
            const float v = sPts[cb][i][m];
            if (g < 2) { if (v < best) { best = v; bi = i; } }
            else       { if (v > best) { best = v; bi = i; } }
        }
        const size_t bboxBase = (size_t)NB * CLS * HWPLANE;
        out[bboxBase + ((size_t)n * 4 + g) * HWPLANE + pbase + m] =
            (g < 2) ? -best : best;

        // gather selected coord's 8 probs, take top-4 descending
        const int cs = 2 * bi + cb;
        float v[BINS];
        #pragma unroll
        for (int b = 0; b < BINS; ++b)
            v[b] = sL[m][CLS + cs * BINS + b];
        #pragma unroll
        for (int k = 0; k < KTOP; ++k) {
            float mx = v[0]; int ib = 0;
            #pragma unroll
            for (int b = 1; b < BINS; ++b)
                if (v[b] > mx) { mx = v[b]; ib = b; }
            sT16[m][g * KTOP + k] = mx;
            #pragma unroll
            for (int b = 0; b < BINS; ++b)
                v[b] = (b == ib) ? -1e30f : v[b];      // cndmask, no dyn index
        }
    }
    __syncthreads();

    // ================= Phase C: 256 threads, (pixel, 8-ch chunk) ===========
    // Each thread's chunk == one GN group, so partials stay thread-local.
    {
        const int m     = tid & 31;
        const int chunk = tid >> 5;           // == GN group, wave-uniform
        float t16l[16];
        #pragma unroll
        for (int i = 0; i < 16; ++i)
            t16l[i] = sT16[m][i];
        float s = 0.f, q = 0.f;
        #pragma unroll
        for (int j = 0; j < CH_PER_G; ++j) {
            const int ch = chunk * CH_PER_G + j;
            float h = conf1_b[ch];
            #pragma unroll
            for (int i = 0; i < 16; ++i)
                h += conf1_w[ch * 16 + i] * t16l[i];
            hid_ws[((size_t)n * CONF_CH + ch) * HWPLANE + pbase + m] = h;
            s += h;
            q += h * h;
        }
        sRed[m][chunk][0] = s;
        sRed[m][chunk][1] = q;
    }
    __syncthreads();

    // block-level reduce of GN partials, one atomic pair per group
    if (tid < GROUPS) {
        float s = 0.f, q = 0.f;
        #pragma unroll
        for (int m = 0; m < TILE_M; ++m) {
            s += sRed[m][tid][0];
            q += sRed[m][tid][1];
        }
        atomicAdd(&part_ws[((size_t)n * GROUPS + tid) * 2 + 0], s);
        atomicAdd(&part_ws[((size_t)n * GROUPS + tid) * 2 + 1], q);
    }
}

// ---------------------------------------------------------------------------
// Kernel B: finalize GroupNorm statistics  (16 (n,g) pairs)
// ---------------------------------------------------------------------------
__global__ void gn_stats_kernel(const float* __restrict__ part,
                                float* __restrict__ stats)
{
    const int i = threadIdx.x;
    if (i < NB * GROUPS) {
        const float cnt  = (float)(CH_PER_G * HWPLANE);
        const float mean = part[i * 2 + 0] / cnt;
        const float var  = part[i * 2 + 1] / cnt - mean * mean;
        stats[i * 2 + 0] = mean;
        stats[i * 2 + 1] = rsqrtf(var + GN_EPS);
    }
}

// ---------------------------------------------------------------------------
// Kernel C: GN + relu + 64->1 conv + sigmoid(conf); cls *= conf in place
// ---------------------------------------------------------------------------
__global__ __launch_bounds__(256) void conf_apply_kernel(
    const float* __restrict__ hid,      // [NB][CONF_CH][HW]
    const float* __restrict__ stats,    // [NB][GROUPS][2]
    const float* __restrict__ gamma,
    const float* __restrict__ beta,
    const float* __restrict__ c2w,      // [CONF_CH]
    const float* __restrict__ c2b,      // [1]
    float* __restrict__ out)
{
    const size_t idx = (size_t)blockIdx.x * blockDim.x + threadIdx.x;
    if (idx >= (size_t)NB * HWPLANE) return;
    const int n = (int)(idx / HWPLANE);
    const int p = (int)(idx % HWPLANE);

    float acc = c2b[0];
    #pragma unroll 8
    for (int ch = 0; ch < CONF_CH; ++ch) {
        const int g = ch >> 3;
        const float mean = stats[((size_t)n * GROUPS + g) * 2 + 0];
        const float inv  = stats[((size_t)n * GROUPS + g) * 2 + 1];
        float h = hid[((size_t)n * CONF_CH + ch) * HWPLANE + p];
        h = (h - mean) * inv * gamma[ch] + beta[ch];
        h = fmaxf(h, 0.f);
        acc += c2w[ch] * h;
    }
    const float conf = sigmoidf(acc);

    float* op = out + (size_t)n * CLS * HWPLANE + p;
    for (int o = 0; o < CLS; ++o)
        op[(size_t)o * HWPLANE] *= conf;
}

// ---------------------------------------------------------------------------
extern "C" void kernel_launch(void* const* d_in, const int* in_sizes, int n_in,
                              void* d_out, int out_size, void* d_ws, size_t ws_size,
                              hipStream_t stream)
{
    const float* cls_feat = (const float*)d_in[0];
    const float* reg_feat = (const float*)d_in[1];
    const float* pts_init = (const float*)d_in[2];
    const float* scale_p  = (const float*)d_in[3];
    const float* cls_w    = (const float*)d_in[4];
    const float* cls_b    = (const float*)d_in[5];
    const float* reg_w    = (const float*)d_in[6];
    const float* reg_b    = (const float*)d_in[7];
    const float* conf1_w  = (const float*)d_in[8];
    const float* conf1_b  = (const float*)d_in[9];
    const float* gn_gamma = (const float*)d_in[10];
    const float* gn_beta  = (const float*)d_in[11];
    const float* conf2_w  = (const float*)d_in[12];
    const float* conf2_b  = (const float*)d_in[13];

    float* out = (float*)d_out;

    // workspace layout: hid (13.1 MB) | GN partials (32 f) | GN stats (32 f)
    const size_t hidElems = (size_t)NB * CONF_CH * HWPLANE;
    float* hid_ws  = (float*)d_ws;
    float* part_ws = hid_ws + hidElems;
    float* stats   = part_ws + NB * GROUPS * 2;

    // zero the atomic accumulators each call (graph-capturable memset node)
    (void)hipMemsetAsync(part_ws, 0, NB * GROUPS * 2 * sizeof(float), stream);

    dim3 gridA(HWPLANE / TILE_M, NB);
    head_gemm_kernel<<<gridA, 256, 0, stream>>>(
        cls_feat, reg_feat, pts_init, scale_p,
        cls_w, cls_b, reg_w, reg_b, conf1_w, conf1_b,
        out, hid_ws, part_ws);

    gn_stats_kernel<<<1, 32, 0, stream>>>(part_ws, stats);

    const int totPix = NB * HWPLANE;
    conf_apply_kernel<<<(totPix + 255) / 256, 256, 0, stream>>>(
        hid_ws, stats, gn_gamma, gn_beta, conf2_w, conf2_b, out);
}